// WelfordEstimator_76656576299291
// MI455X (gfx1250) — compile-verified
//
#include <hip/hip_runtime.h>
#include <stdint.h>

// Problem shape (compile-time, from the reference)
constexpr int Bn   = 64;                 // batch (reduction) dim
constexpr int CHW  = 64 * 128 * 128;     // 1,048,576 locations
constexpr int TPB  = 256;                // 8 wave32 per block
constexpr int VEC  = 4;                  // float4 per thread
constexpr int LPB  = TPB * VEC;          // 1024 locations per block
constexpr int STAGE = 4;                 // batch steps per pipeline stage
constexpr int NSTG  = Bn / STAGE;        // 16 stages
// LDS: 2 buffers x STAGE steps x LPB floats = 32 KB per workgroup

typedef float f32x4 __attribute__((ext_vector_type(4)));
typedef int   i32x4 __attribute__((ext_vector_type(4)));

// Pointer flavors for the async-copy builtin: payload-typed (v4i32) pointers.
// AS-qualified so they match either generic or qualified builtin signatures
// (specific -> generic converts implicitly; the reverse would diagnose).
typedef __attribute__((address_space(1))) i32x4 g_i32x4;
typedef __attribute__((address_space(3))) i32x4 l_i32x4;

// ---- CDNA5 async global -> LDS copy (ASYNCcnt-tracked) --------------------
__device__ __forceinline__ void async_ld_b128(const float* g, float* l) {
#if __has_builtin(__builtin_amdgcn_global_load_async_to_lds_b128)
  __builtin_amdgcn_global_load_async_to_lds_b128(
      (g_i32x4*)g,            // global source (VADDR)
      (l_i32x4*)l,            // LDS destination (VDST)
      /*imm offset*/0, /*cpol*/0);
#else
  unsigned loff =
      (unsigned)(unsigned long)(__attribute__((address_space(3))) void*)l;
  asm volatile("global_load_async_to_lds_b128 %0, %1, off"
               :: "v"(loff), "v"(g) : "memory");
#endif
}

// s_wait_asynccnt with an immediate; memory clobber fences the LDS reads.
#define WAIT_ASYNC(N) asm volatile("s_wait_asynccnt %0" :: "i"(N) : "memory")

__global__ void __launch_bounds__(TPB)
welford_kernel(const float* __restrict__ x,
               const float* __restrict__ m0,
               const float* __restrict__ s0,
               const int*   __restrict__ n0,
               float*       __restrict__ out)
{
  __shared__ float lds[2 * STAGE * LPB];   // 32 KB, double-buffered stages

  const int tid = threadIdx.x;
  const int loc = blockIdx.x * LPB + tid * VEC;   // 16B-aligned float4 slot

  const float nbase = (float)n0[0];

  f32x4 m = *(const f32x4*)(m0 + loc);
  f32x4 s = *(const f32x4*)(s0 + loc);

  const float* gbase = x + loc;
  float*       lbase = &lds[tid * VEC];

  // Issue stage 0 into buffer 0.
#pragma unroll
  for (int j = 0; j < STAGE; ++j)
    async_ld_b128(gbase + (size_t)j * CHW, lbase + j * LPB);

  for (int st = 0; st < NSTG; ++st) {
    const int buf = st & 1;

    if (st + 1 < NSTG) {
      // Prefetch next stage into the other buffer, then wait until only
      // those STAGE loads remain outstanding (async loads retire in order).
      const float* gn = gbase + (size_t)(st + 1) * STAGE * CHW;
      float*       ln = lbase + (buf ^ 1) * STAGE * LPB;
#pragma unroll
      for (int j = 0; j < STAGE; ++j)
        async_ld_b128(gn + (size_t)j * CHW, ln + j * LPB);
      WAIT_ASYNC(STAGE);
    } else {
      WAIT_ASYNC(0);
    }

    // Consume current stage from LDS: 4-lane Welford recurrence per step.
#pragma unroll
    for (int j = 0; j < STAGE; ++j) {
      const int b = st * STAGE + j;
      // Uniform per-step 1/(n+1): single v_rcp_f32, no IEEE div sequence.
      const float rinv = __builtin_amdgcn_rcpf(nbase + (float)(b + 1));

      f32x4 xi = *(const f32x4*)&lds[(buf * STAGE + j) * LPB + tid * VEC];
      f32x4 d  = xi - m;        // xi - m_old
      m        = m + d * rinv;  // m_new = m + (xi - m_old)/(n+1)
      f32x4 d2 = xi - m;        // xi - m_new
      s        = s + d2 * d;    // s += (xi - m_new)*(xi - m_old)
    }
  }

  // out[0:CHW) = m, out[CHW:2CHW) = s   (both B128 coalesced stores)
  *(f32x4*)(out + loc)        = m;
  *(f32x4*)(out + CHW + loc)  = s;
}

extern "C" void kernel_launch(void* const* d_in, const int* in_sizes, int n_in,
                              void* d_out, int out_size, void* d_ws, size_t ws_size,
                              hipStream_t stream) {
  const float* x  = (const float*)d_in[0];
  const float* m0 = (const float*)d_in[1];
  const float* s0 = (const float*)d_in[2];
  const int*   n0 = (const int*)d_in[3];
  float* out = (float*)d_out;

  dim3 grid(CHW / LPB);   // 1024 blocks
  welford_kernel<<<grid, TPB, 0, stream>>>(x, m0, s0, n0, out);
}